// FPModule_16870631538822
// MI455X (gfx1250) — compile-verified
//
#include <hip/hip_runtime.h>

typedef __attribute__((ext_vector_type(2))) float v2f;
typedef __attribute__((ext_vector_type(8))) float v8f;

#define BATCHES   16
#define NCOARSE   1024
#define MFINE     4096
#define CFEAT     256
#define CSKIP     128
#define KIN       384     // CFEAT + CSKIP
#define HID       256
#define ROWS      32      // rows (fine points) per MLP workgroup: 2 row tiles
#define EPSD      1e-16f

// ---------------------------------------------------------------------------
// Kernel 1: per-batch KNN (K=3) + inverse-distance interpolation.
// Block = 256 threads handles 256 consecutive fine points (same cloud).
// Writes y[65536,256] into d_out (staging; the MLP kernel reads only its own
// row block before overwriting it).
// ---------------------------------------------------------------------------
__global__ __launch_bounds__(256)
void knn_interp_kernel(const float* __restrict__ x,
                       const float* __restrict__ pos,
                       const float* __restrict__ pos_skip,
                       float* __restrict__ y)
{
    __shared__ float sPx[NCOARSE];
    __shared__ float sPy[NCOARSE];
    __shared__ float sPz[NCOARSE];
    __shared__ int   sIdx[256 * 3];
    __shared__ float sW[256 * 3];

    const int tid   = threadIdx.x;
    const int batch = blockIdx.x >> 4;          // 16 blocks per cloud
    const int coarseBase = batch * NCOARSE;

    for (int j = tid; j < NCOARSE; j += 256) {
        const float* p = pos + (size_t)(coarseBase + j) * 3;
        sPx[j] = p[0];
        sPy[j] = p[1];
        sPz[j] = p[2];
    }
    __syncthreads();

    const int fp = blockIdx.x * 256 + tid;
    const float qx = pos_skip[(size_t)fp * 3 + 0];
    const float qy = pos_skip[(size_t)fp * 3 + 1];
    const float qz = pos_skip[(size_t)fp * 3 + 2];

    float d0 = 3.4e38f, d1 = 3.4e38f, d2 = 3.4e38f;
    int   i0 = 0, i1 = 0, i2 = 0;
    for (int j = 0; j < NCOARSE; ++j) {
        const float dx = qx - sPx[j];
        const float dy = qy - sPy[j];
        const float dz = qz - sPz[j];
        const float d  = dx * dx + dy * dy + dz * dz;
        if (d < d2) {
            if (d < d1) {
                d2 = d1; i2 = i1;
                if (d < d0) { d1 = d0; i1 = i0; d0 = d; i0 = j; }
                else        { d1 = d;  i1 = j; }
            } else {
                d2 = d; i2 = j;
            }
        }
    }
    const float w0 = 1.0f / fmaxf(d0, EPSD);
    const float w1 = 1.0f / fmaxf(d1, EPSD);
    const float w2 = 1.0f / fmaxf(d2, EPSD);
    const float inv = 1.0f / (w0 + w1 + w2);
    sIdx[tid * 3 + 0] = i0;  sW[tid * 3 + 0] = w0 * inv;
    sIdx[tid * 3 + 1] = i1;  sW[tid * 3 + 1] = w1 * inv;
    sIdx[tid * 3 + 2] = i2;  sW[tid * 3 + 2] = w2 * inv;
    __syncthreads();

    // Role swap: thread = channel. Coalesced 1KB row gathers (x is L2-resident).
    const size_t outBase = (size_t)blockIdx.x * 256 * CFEAT;
    for (int p = 0; p < 256; ++p) {
        const int   j0 = sIdx[p * 3 + 0];
        const int   j1 = sIdx[p * 3 + 1];
        const int   j2 = sIdx[p * 3 + 2];
        const float a0 = sW[p * 3 + 0];
        const float a1 = sW[p * 3 + 1];
        const float a2 = sW[p * 3 + 2];
        const float* r0 = x + (size_t)(coarseBase + j0) * CFEAT;
        const float* r1 = x + (size_t)(coarseBase + j1) * CFEAT;
        const float* r2 = x + (size_t)(coarseBase + j2) * CFEAT;
        y[outBase + (size_t)p * CFEAT + tid] =
            a0 * r0[tid] + a1 * r1[tid] + a2 * r2[tid];
    }
}

// ---------------------------------------------------------------------------
// Kernel 2: fused 2-layer MLP with V_WMMA_F32_16X16X4_F32.
// Block = 256 threads (8 waves), 32 rows per block. Wave w owns a 2x2 grid of
// 16x16 tiles: rows {0..15, 16..31} x cols [32w, 32w+32). One B-fragment pair
// now feeds 4 WMMAs (2x arithmetic intensity on L2-resident W1/W2).
// ---------------------------------------------------------------------------
__global__ __launch_bounds__(256)
void fp_mlp_kernel(const float* __restrict__ y,      // aliases d_out
                   const float* __restrict__ x_skip,
                   const float* __restrict__ W1,
                   const float* __restrict__ b1,
                   const float* __restrict__ W2,
                   const float* __restrict__ b2,
                   float* __restrict__ out)
{
    __shared__ float A1[ROWS][KIN + 4];   // stride 388: 388 % 64 == 4 -> conflict-free
    __shared__ float H1[ROWS][HID + 4];   // stride 260: 260 % 64 == 4

    const int tid  = threadIdx.x;
    const int lane = tid & 31;
    const int wave = tid >> 5;
    const int lo   = lane & 15;
    const int hi   = lane >> 4;
    const int rowBase = blockIdx.x * ROWS;

    // ---- stage A = concat(y_rows, x_skip_rows) : 32 x 384 ----
    for (int t = tid; t < ROWS * CFEAT; t += 256) {
        const int r = t >> 8, c = t & (CFEAT - 1);
        A1[r][c] = y[(size_t)(rowBase + r) * CFEAT + c];
    }
    for (int t = tid; t < ROWS * CSKIP; t += 256) {
        const int r = t >> 7, c = t & (CSKIP - 1);
        A1[r][CFEAT + c] = x_skip[(size_t)(rowBase + r) * CSKIP + c];
    }
    __syncthreads();

    // ---- GEMM1: h1 = relu(A @ W1 + b1), K = 384 ----
    const int n0 = wave * 32;
    v8f acc00 = {0.f,0.f,0.f,0.f,0.f,0.f,0.f,0.f};   // rowTile0 x colTile0
    v8f acc01 = {0.f,0.f,0.f,0.f,0.f,0.f,0.f,0.f};   // rowTile0 x colTile1
    v8f acc10 = {0.f,0.f,0.f,0.f,0.f,0.f,0.f,0.f};   // rowTile1 x colTile0
    v8f acc11 = {0.f,0.f,0.f,0.f,0.f,0.f,0.f,0.f};   // rowTile1 x colTile1
    #pragma unroll 8
    for (int k0 = 0; k0 < KIN; k0 += 4) {
        const int kr = k0 + 2 * hi;
        v2f a0; a0.x = A1[lo][kr];
                a0.y = A1[lo][kr + 1];
        v2f a1; a1.x = A1[16 + lo][kr];
                a1.y = A1[16 + lo][kr + 1];
        v2f bA; bA.x = W1[(size_t)kr * HID + n0 + lo];
                bA.y = W1[(size_t)(kr + 1) * HID + n0 + lo];
        v2f bB; bB.x = W1[(size_t)kr * HID + n0 + 16 + lo];
                bB.y = W1[(size_t)(kr + 1) * HID + n0 + 16 + lo];
        acc00 = __builtin_amdgcn_wmma_f32_16x16x4_f32(false, a0, false, bA,
                                                      (short)0, acc00, false, false);
        acc01 = __builtin_amdgcn_wmma_f32_16x16x4_f32(false, a0, false, bB,
                                                      (short)0, acc01, false, false);
        acc10 = __builtin_amdgcn_wmma_f32_16x16x4_f32(false, a1, false, bA,
                                                      (short)0, acc10, false, false);
        acc11 = __builtin_amdgcn_wmma_f32_16x16x4_f32(false, a1, false, bB,
                                                      (short)0, acc11, false, false);
    }
    {
        const float bias0 = b1[n0 + lo];
        const float bias1 = b1[n0 + 16 + lo];
        #pragma unroll
        for (int r = 0; r < 8; ++r) {
            float v00 = acc00[r] + bias0; v00 = v00 > 0.f ? v00 : 0.f;
            float v01 = acc01[r] + bias1; v01 = v01 > 0.f ? v01 : 0.f;
            float v10 = acc10[r] + bias0; v10 = v10 > 0.f ? v10 : 0.f;
            float v11 = acc11[r] + bias1; v11 = v11 > 0.f ? v11 : 0.f;
            H1[r + 8 * hi][n0 + lo]           = v00;
            H1[r + 8 * hi][n0 + 16 + lo]      = v01;
            H1[16 + r + 8 * hi][n0 + lo]      = v10;
            H1[16 + r + 8 * hi][n0 + 16 + lo] = v11;
        }
    }
    __syncthreads();

    // ---- GEMM2: out = relu(h1 @ W2 + b2), K = 256 ----
    v8f acc20 = {0.f,0.f,0.f,0.f,0.f,0.f,0.f,0.f};
    v8f acc21 = {0.f,0.f,0.f,0.f,0.f,0.f,0.f,0.f};
    v8f acc30 = {0.f,0.f,0.f,0.f,0.f,0.f,0.f,0.f};
    v8f acc31 = {0.f,0.f,0.f,0.f,0.f,0.f,0.f,0.f};
    #pragma unroll 8
    for (int k0 = 0; k0 < HID; k0 += 4) {
        const int kr = k0 + 2 * hi;
        v2f a0; a0.x = H1[lo][kr];
                a0.y = H1[lo][kr + 1];
        v2f a1; a1.x = H1[16 + lo][kr];
                a1.y = H1[16 + lo][kr + 1];
        v2f bA; bA.x = W2[(size_t)kr * HID + n0 + lo];
                bA.y = W2[(size_t)(kr + 1) * HID + n0 + lo];
        v2f bB; bB.x = W2[(size_t)kr * HID + n0 + 16 + lo];
                bB.y = W2[(size_t)(kr + 1) * HID + n0 + 16 + lo];
        acc20 = __builtin_amdgcn_wmma_f32_16x16x4_f32(false, a0, false, bA,
                                                      (short)0, acc20, false, false);
        acc21 = __builtin_amdgcn_wmma_f32_16x16x4_f32(false, a0, false, bB,
                                                      (short)0, acc21, false, false);
        acc30 = __builtin_amdgcn_wmma_f32_16x16x4_f32(false, a1, false, bA,
                                                      (short)0, acc30, false, false);
        acc31 = __builtin_amdgcn_wmma_f32_16x16x4_f32(false, a1, false, bB,
                                                      (short)0, acc31, false, false);
    }
    {
        const float bias0 = b2[n0 + lo];
        const float bias1 = b2[n0 + 16 + lo];
        #pragma unroll
        for (int r = 0; r < 8; ++r) {
            float v00 = acc20[r] + bias0; v00 = v00 > 0.f ? v00 : 0.f;
            float v01 = acc21[r] + bias1; v01 = v01 > 0.f ? v01 : 0.f;
            float v10 = acc30[r] + bias0; v10 = v10 > 0.f ? v10 : 0.f;
            float v11 = acc31[r] + bias1; v11 = v11 > 0.f ? v11 : 0.f;
            const size_t orow0 = (size_t)(rowBase + r + 8 * hi) * HID;
            const size_t orow1 = (size_t)(rowBase + 16 + r + 8 * hi) * HID;
            out[orow0 + n0 + lo]      = v00;
            out[orow0 + n0 + 16 + lo] = v01;
            out[orow1 + n0 + lo]      = v10;
            out[orow1 + n0 + 16 + lo] = v11;
        }
    }
}

// ---------------------------------------------------------------------------
extern "C" void kernel_launch(void* const* d_in, const int* in_sizes, int n_in,
                              void* d_out, int out_size, void* d_ws, size_t ws_size,
                              hipStream_t stream)
{
    // setup_inputs order: x, pos, batch, x_skip, pos_skip, batch_skip, W1, b1, W2, b2
    const float* x        = (const float*)d_in[0];
    const float* pos      = (const float*)d_in[1];
    const float* x_skip   = (const float*)d_in[3];
    const float* pos_skip = (const float*)d_in[4];
    const float* W1       = (const float*)d_in[6];
    const float* b1       = (const float*)d_in[7];
    const float* W2       = (const float*)d_in[8];
    const float* b2       = (const float*)d_in[9];
    float* out = (float*)d_out;

    // Stage y (knn-interpolated features) directly in d_out: each MLP block
    // reads only its own 32 rows of y before overwriting those same rows.
    knn_interp_kernel<<<dim3((BATCHES * MFINE) / 256), dim3(256), 0, stream>>>(
        x, pos, pos_skip, out);

    fp_mlp_kernel<<<dim3((BATCHES * MFINE) / ROWS), dim3(256), 0, stream>>>(
        out, x_skip, W1, b1, W2, b2, out);
}